// KalmanFilter_69466801045752
// MI455X (gfx1250) — compile-verified
//
#include <hip/hip_runtime.h>
#include <hip/hip_bf16.h>
#include <math.h>

// ---------------------------------------------------------------------------
// Kalman filter + RTS smoother for MI455X (gfx1250, wave32, WMMA).
//
// M=32 states, K=64 obs, NOBS=8192 steps.  The forward filter and the
// backward smoother are serial recursions -> one persistent 256-thread
// (8-wave) workgroup each, with ALL per-step state in LDS (~61 KB of the
// 320 KB WGP LDS).  The smoother gains J[t] are independent across t ->
// 8192 parallel workgroups.  Every dense GEMM (32x32x32 .. 64x64x32) is
// done with V_WMMA_F32_16X16X4_F32 tiles (f32 accumulate matches the
// reference precision; lower-precision WMMA would destabilize the
// Cholesky recursion).  Total HBM traffic ~170 MB << 23.3 TB/s * runtime,
// and it all fits in the 192 MB L2 -- this workload is latency-bound, so
// the win is keeping the recursion state LDS-resident and feeding the
// matrix pipes with WMMA instead of scalar VALU loops.
//
// Eigen-clip note: Q = qq^T/M + 0.1 I and V0 = vv^T/M + 0.1 I have
// min-eigenvalue >= 0.1 >> MIN_EIG=1e-8, so _cov_stable == symmetrize.
// ---------------------------------------------------------------------------

#define NOBS 8192
#define MDIM 32
#define KDIM 64

typedef __attribute__((ext_vector_type(2))) float v2f;
typedef __attribute__((ext_vector_type(8))) float v8f;

// ---- workspace layout (float offsets) -------------------------------------
// total = 25,695,296 floats = ~98 MB (d_ws must be at least this big)
#define WS_ZM   0                               // NOBS*32   prior means Z_{t|t-1}
#define WS_VM   (WS_ZM  + NOBS*32)              // NOBS*1024 prior covs  V_{t|t-1}
#define WS_ZU   (WS_VM  + NOBS*1024)            // NOBS*32   posterior means
#define WS_VU   (WS_ZU  + NOBS*32)              // NOBS*1024 posterior covs
#define WS_J    (WS_VU  + NOBS*1024)            // NOBS*1024 smoother gains
#define WS_KG   (WS_J   + NOBS*1024)            // 2048      last Kalman gain (32x64)
#define WS_AQ   (WS_KG  + 2048)                 // 1024      clipped A
#define WS_QS   (WS_AQ  + 1024)                 // 1024      symmetrized Q
#define WS_V0S  (WS_QS  + 1024)                 // 1024      symmetrized V0
#define WS_RD   (WS_V0S + 1024)                 // 64        clamped diag(R)

// ---- output layout (float offsets into d_out) ------------------------------
#define OUT_ZMT   0                             // (32, NOBS+1)
#define OUT_VMT   (32*(NOBS+1))                 // (32,32, NOBS+1)
#define OUT_VMT1  (OUT_VMT + 1024*(NOBS+1))     // (32,32, NOBS)
#define OUT_LL    (OUT_VMT1 + 1024*NOBS)        // scalar

// ---------------------------------------------------------------------------
// Generic LDS-resident tiled GEMM:  D(16*Mt x 16*Nt) = opA(A) * opB(B)
// using V_WMMA_F32_16X16X4_F32.  Tiles distributed across waves; every
// branch is wave-uniform so EXEC is all-ones at each WMMA.
// A-fragment (16x4 f32): v0 = K{0,1} for lanes 0-15 / K{2,3}? -> per ISA:
//   lanes 0-15 hold K = k0,k0+1 (VGPR0,VGPR1); lanes 16-31 hold K = k0+2,k0+3.
// C/D (16x16 f32): VGPR r: lanes 0-15 -> M=r, lanes 16-31 -> M=r+8.
// ---------------------------------------------------------------------------
template <bool TA, bool TB>
__device__ inline void lds_gemm(const float* __restrict__ A, int lda,
                                const float* __restrict__ B, int ldb,
                                float* __restrict__ D, int ldd,
                                int Mtiles, int Ntiles, int K)
{
  const int lane = threadIdx.x & 31;
  const int wave = threadIdx.x >> 5;
  const int nw   = blockDim.x >> 5;
  const int half = lane >> 4;      // 0: lanes 0-15, 1: lanes 16-31
  const int l    = lane & 15;
  const int ntiles = Mtiles * Ntiles;
  for (int tidx = wave; tidx < ntiles; tidx += nw) {
    const int tm = (tidx / Ntiles) * 16;
    const int tn = (tidx % Ntiles) * 16;
    v8f acc = {0.f, 0.f, 0.f, 0.f, 0.f, 0.f, 0.f, 0.f};
    for (int k0 = 0; k0 < K; k0 += 4) {
      v2f a, b;
      const int ka = k0 + half * 2;
      if (TA) { a.x = A[ka * lda + (tm + l)]; a.y = A[(ka + 1) * lda + (tm + l)]; }
      else    { a.x = A[(tm + l) * lda + ka]; a.y = A[(tm + l) * lda + ka + 1]; }
      const int kb = k0 + half;
      if (TB) { b.x = B[(tn + l) * ldb + kb]; b.y = B[(tn + l) * ldb + kb + 2]; }
      else    { b.x = B[kb * ldb + (tn + l)]; b.y = B[(kb + 2) * ldb + (tn + l)]; }
      acc = __builtin_amdgcn_wmma_f32_16x16x4_f32(
          /*neg_a=*/false, a, /*neg_b=*/false, b,
          /*c_mod=*/(short)0, acc, /*reuse_a=*/false, /*reuse_b=*/false);
    }
#pragma unroll
    for (int r = 0; r < 8; ++r)
      D[(tm + r + half * 8) * ldd + (tn + l)] = acc[r];
  }
}

// ---- block-cooperative Cholesky (in-place, lower triangle of F, n<=64) -----
__device__ inline void lds_cholesky(float* F, int n)
{
  const int tid = threadIdx.x;
  for (int j = 0; j < n; ++j) {
    if (tid == 0) F[j * n + j] = sqrtf(fmaxf(F[j * n + j], 1e-30f));
    __syncthreads();
    const float d = F[j * n + j];
    if (tid > j && tid < n) F[tid * n + j] /= d;
    __syncthreads();
    for (int e = tid; e < n * n; e += blockDim.x) {
      const int i = e / n, k2 = e % n;
      if (i > j && k2 > j && k2 <= i) F[i * n + k2] -= F[i * n + j] * F[k2 * n + j];
    }
    __syncthreads();
  }
}

// ---- cho_solve: F X = B via L (lower). One column per thread. --------------
__device__ inline void lds_chol_solve(const float* L, int n,
                                      const float* Bm, int ldb,
                                      float* X, int ldx, int ncols)
{
  const int c = threadIdx.x;
  if (c < ncols) {
    for (int i = 0; i < n; ++i) {                 // L y = b
      float s = Bm[i * ldb + c];
      for (int j = 0; j < i; ++j) s -= L[i * n + j] * X[j * ldx + c];
      X[i * ldx + c] = s / L[i * n + i];
    }
    for (int i = n - 1; i >= 0; --i) {            // L^T x = y
      float s = X[i * ldx + c];
      for (int j = i + 1; j < n; ++j) s -= L[j * n + i] * X[j * ldx + c];
      X[i * ldx + c] = s / L[i * n + i];
    }
  }
  __syncthreads();
}

__device__ inline void g2s(float* dst, const float* src, int n)
{
  for (int i = threadIdx.x; i < n; i += blockDim.x) dst[i] = src[i];
}

// ===========================================================================
// Kernel 0: preprocessing (symmetrize Q/V0, clip A, clamp diag R)
// ===========================================================================
__global__ __launch_bounds__(256) void kf_preprocess(
    const float* __restrict__ A, const float* __restrict__ Q,
    const float* __restrict__ R, const float* __restrict__ V0,
    float* __restrict__ ws)
{
  const int tid = threadIdx.x;
  for (int e = tid; e < 1024; e += blockDim.x) {
    const int i = e >> 5, j = e & 31;
    ws[WS_AQ + e]  = fminf(0.99f, fmaxf(-0.99f, A[e]));
    ws[WS_QS + e]  = 0.5f * (Q[i * 32 + j] + Q[j * 32 + i]);   // eig-clip is no-op (PD)
    ws[WS_V0S + e] = 0.5f * (V0[i * 32 + j] + V0[j * 32 + i]);
  }
  for (int k2 = tid; k2 < 64; k2 += blockDim.x)
    ws[WS_RD + k2] = fmaxf(R[k2 * 64 + k2], 1e-6f);
}

// ===========================================================================
// Kernel 1: forward Kalman filter (single persistent workgroup, 8 waves)
// ===========================================================================
__global__ __launch_bounds__(256) void kf_forward(
    const float* __restrict__ Y,   // (64, NOBS)
    const float* __restrict__ Cg,  // (64, 32)
    const float* __restrict__ Z0,  // (32,)
    float* __restrict__ ws, float* __restrict__ out)
{
  __shared__ float sA[1024], sC[2048], sQ[1024], sRd[64];
  __shared__ float sZu[32], sVu[1024];
  __shared__ float sZ[32], sT[1024], sV[1024];
  __shared__ float sCVI[64 * 33];   // cols 0..31 = C@V, col 32 = innovation
  __shared__ float sF[64 * 64];     // F, then its Cholesky factor (lower)
  __shared__ float sX[64 * 33];     // F^{-1} [C@V | innov]
  const int tid = threadIdx.x;

  g2s(sA, ws + WS_AQ, 1024);
  g2s(sC, Cg, 2048);
  g2s(sQ, ws + WS_QS, 1024);
  if (tid < 64) sRd[tid] = ws[WS_RD + tid];
  if (tid < 32) sZu[tid] = Z0[tid];
  g2s(sVu, ws + WS_V0S, 1024);
  __syncthreads();

  float ll = 0.f;                       // meaningful on thread 0
  const float log2pi = 1.8378770664093453f;

  for (int t = 0; t < NOBS; ++t) {
    // ---- prior mean Z = A Zu (32 threads) + T = A Vu (WMMA) ----
    if (tid < 32) {
      float s = 0.f;
      for (int k2 = 0; k2 < 32; ++k2) s += sA[tid * 32 + k2] * sZu[k2];
      sZ[tid] = s;
    }
    lds_gemm<false, false>(sA, 32, sVu, 32, sT, 32, 2, 2, 32);
    __syncthreads();
    // ---- prior cov V = sym(T A^T + Q) ----
    lds_gemm<false, true>(sT, 32, sA, 32, sV, 32, 2, 2, 32);
    __syncthreads();
    for (int e = tid; e < 1024; e += blockDim.x) {
      const int i = e >> 5, j = e & 31;
      if (i <= j) {
        const float s = 0.5f * (sV[i * 32 + j] + sV[j * 32 + i]) + sQ[i * 32 + j];
        sV[i * 32 + j] = s; sV[j * 32 + i] = s;
      }
    }
    __syncthreads();
    // ---- innovation (64 threads) + CV = C V (WMMA, into cols 0..31) ----
    if (tid < 64) {
      float s = Y[tid * NOBS + t];
      for (int k2 = 0; k2 < 32; ++k2) s -= sC[tid * 32 + k2] * sZ[k2];
      sCVI[tid * 33 + 32] = s;
    }
    lds_gemm<false, false>(sC, 32, sV, 32, sCVI, 33, 4, 2, 32);
    __syncthreads();
    // ---- F = sym(CV C^T) + diag(Rd) + reg I ----
    lds_gemm<false, true>(sCVI, 33, sC, 32, sF, 64, 4, 4, 32);
    __syncthreads();
    for (int e = tid; e < 4096; e += blockDim.x) {
      const int i = e >> 6, j = e & 63;
      if (i <= j) {
        float s = 0.5f * (sF[i * 64 + j] + sF[j * 64 + i]);
        if (i == j) s += sRd[i] + 1e-6f;
        sF[i * 64 + j] = s; sF[j * 64 + i] = s;
      }
    }
    __syncthreads();
    // ---- Cholesky + solves: X = F^{-1} [CV | innov] ----
    lds_cholesky(sF, 64);
    lds_chol_solve(sF, 64, sCVI, 33, sX, 33, 33);
    // ---- posterior mean: Zu2 = Z + CV^T (F^{-1} innov) ----
    if (tid < 32) {
      float s = sZ[tid];
      for (int k2 = 0; k2 < 64; ++k2) s += sCVI[k2 * 33 + tid] * sX[k2 * 33 + 32];
      sZu[tid] = s;                    // old sZu fully consumed above
    }
    // ---- posterior cov: Vu2 = sym(V - X^T CV)  (Kg = X^T) ----
    lds_gemm<true, false>(sX, 33, sCVI, 33, sT, 32, 2, 2, 64);
    __syncthreads();
    for (int e = tid; e < 1024; e += blockDim.x) {
      const int i = e >> 5, j = e & 31;
      if (i <= j) {
        const float s = sV[i * 32 + j] - 0.5f * (sT[i * 32 + j] + sT[j * 32 + i]);
        sVu[i * 32 + j] = s; sVu[j * 32 + i] = s;
      }
    }
    // ---- log-likelihood (thread 0, serial over 64) ----
    if (tid == 0) {
      float ld = 0.f, q = 0.f;
      for (int i = 0; i < 64; ++i) {
        ld += logf(sF[i * 64 + i]);
        q  += sCVI[i * 33 + 32] * sX[i * 33 + 32];
      }
      ll += -0.5f * (64.f * log2pi + 2.f * ld + q);
    }
    __syncthreads();
    // ---- persist this step ----
    if (tid < 32) {
      ws[WS_ZM + t * 32 + tid] = sZ[tid];
      ws[WS_ZU + t * 32 + tid] = sZu[tid];
    }
    for (int e = tid; e < 1024; e += blockDim.x) {
      ws[WS_VM + t * 1024 + e] = sV[e];
      ws[WS_VU + t * 1024 + e] = sVu[e];
    }
    if (t == NOBS - 1)
      for (int e = tid; e < 2048; e += blockDim.x) {
        const int m = e >> 6, k2 = e & 63;
        ws[WS_KG + e] = sX[k2 * 33 + m];          // Kg = X^T (32x64)
      }
    __syncthreads();
  }

  // tail of outputs: ZmT[:,N] = ZmU[-1], VmT[:,:,N] = VmU[-1], loglik
  if (tid < 32) out[OUT_ZMT + tid * (NOBS + 1) + NOBS] = sZu[tid];
  for (int e = tid; e < 1024; e += blockDim.x)
    out[OUT_VMT + e * (NOBS + 1) + NOBS] = sVu[e];
  if (tid == 0) out[OUT_LL] = __builtin_isfinite(ll) ? ll : -__builtin_huge_valf();
}

// ===========================================================================
// Kernel 2: smoother gains J[t] = (solve(Vm[t]+reg I, A VmU[t]))^T, parallel
// ===========================================================================
__global__ __launch_bounds__(128) void kf_gain(float* __restrict__ ws)
{
  const int t = blockIdx.x;
  const int tid = threadIdx.x;
  __shared__ float sA[1024], sVp[1024], sG[1024], sB[1024], sX[32 * 33];

  g2s(sA, ws + WS_AQ, 1024);
  g2s(sVp, (t == 0) ? (ws + WS_V0S) : (ws + WS_VU + (t - 1) * 1024), 1024);
  for (int e = tid; e < 1024; e += blockDim.x) {
    const int i = e >> 5, j = e & 31;
    sG[e] = ws[WS_VM + t * 1024 + e] + ((i == j) ? 1e-6f : 0.f);
  }
  __syncthreads();
  lds_gemm<false, false>(sA, 32, sVp, 32, sB, 32, 2, 2, 32);   // B = A VmU[t]
  __syncthreads();
  lds_cholesky(sG, 32);
  lds_chol_solve(sG, 32, sB, 32, sX, 33, 32);                  // X = G^{-1} B
  for (int e = tid; e < 1024; e += blockDim.x) {
    const int i = e >> 5, j = e & 31;
    ws[WS_J + t * 1024 + e] = sX[j * 33 + i];                  // J = X^T
  }
}

// ===========================================================================
// Kernel 3: fused backward RTS smoother + lag-1 covariance (one workgroup)
// ===========================================================================
__global__ __launch_bounds__(256) void kf_backward(
    const float* __restrict__ Cg, const float* __restrict__ Z0,
    float* __restrict__ ws, float* __restrict__ out)
{
  __shared__ float sA[1024], sC[2048], sKg[2048];
  __shared__ float sJa[1024], sJb[1024];
  __shared__ float sVmU[1024], sVm[1024], sVnext[1024], sVV[1024];
  __shared__ float sT1[1024], sT2[1024], sVt[1024];
  __shared__ float szn[32], szt[32];
  const int tid = threadIdx.x;

  g2s(sA, ws + WS_AQ, 1024);
  g2s(sC, Cg, 2048);
  g2s(sKg, ws + WS_KG, 2048);
  g2s(sVnext, ws + WS_VU + (NOBS - 1) * 1024, 1024);   // V_next = VmU[-1]
  if (tid < 32) szn[tid] = ws[WS_ZU + (NOBS - 1) * 32 + tid];
  g2s(sVmU, ws + WS_VU + (NOBS - 2) * 1024, 1024);     // Vu[N-2]
  __syncthreads();

  // VV_T = (I - Kg C) A Vu[N-2]
  lds_gemm<false, false>(sA, 32, sVmU, 32, sT1, 32, 2, 2, 32);   // A Vu[N-2]
  __syncthreads();
  lds_gemm<false, false>(sKg, 64, sC, 32, sT2, 32, 2, 2, 64);    // Kg C
  __syncthreads();
  lds_gemm<false, false>(sT2, 32, sT1, 32, sVV, 32, 2, 2, 32);   // (Kg C)(A Vu)
  __syncthreads();
  for (int e = tid; e < 1024; e += blockDim.x) sVV[e] = sT1[e] - sVV[e];
  __syncthreads();
  for (int e = tid; e < 1024; e += blockDim.x)
    out[OUT_VMT1 + e * NOBS + (NOBS - 1)] = sVV[e];

  g2s(sJa, ws + WS_J + (NOBS - 1) * 1024, 1024);
  __syncthreads();
  float* Jt = sJa;
  float* Jp = sJb;

  for (int t = NOBS - 1; t >= 0; --t) {
    if (t > 0) g2s(Jp, ws + WS_J + (t - 1) * 1024, 1024);
    g2s(sVmU, (t == 0) ? (ws + WS_V0S) : (ws + WS_VU + (t - 1) * 1024), 1024);
    g2s(sVm, ws + WS_VM + t * 1024, 1024);
    __syncthreads();

    // ---- smoother: V_t = sym(VmU + J (Vnext - Vm) J^T) ----
    for (int e = tid; e < 1024; e += blockDim.x) sT1[e] = sVnext[e] - sVm[e];
    __syncthreads();
    lds_gemm<false, false>(Jt, 32, sT1, 32, sT2, 32, 2, 2, 32);
    __syncthreads();
    lds_gemm<false, true>(sT2, 32, Jt, 32, sT1, 32, 2, 2, 32);
    __syncthreads();
    if (tid < 32) {                              // z_t = ZmU + J (z_next - Zm)
      const float zmU = (t == 0) ? Z0[tid] : ws[WS_ZU + (t - 1) * 32 + tid];
      float s = zmU;
      for (int k2 = 0; k2 < 32; ++k2)
        s += Jt[tid * 32 + k2] * (szn[k2] - ws[WS_ZM + t * 32 + k2]);
      szt[tid] = s;
    }
    for (int e = tid; e < 1024; e += blockDim.x) {
      const int i = e >> 5, j = e & 31;
      if (i <= j) {
        const float s = sVmU[i * 32 + j] + 0.5f * (sT1[i * 32 + j] + sT1[j * 32 + i]);
        sVt[i * 32 + j] = s; sVt[j * 32 + i] = s;
      }
    }
    __syncthreads();
    if (tid < 32) out[OUT_ZMT + tid * (NOBS + 1) + t] = szt[tid];
    for (int e = tid; e < 1024; e += blockDim.x)
      out[OUT_VMT + e * (NOBS + 1) + t] = sVt[e];

    // ---- lag-1: VV_{t-1} = (VmU_t + J_t (VV_t - A VmU_t)) J_{t-1}^T ----
    if (t >= 1) {
      lds_gemm<false, false>(sA, 32, sVmU, 32, sT1, 32, 2, 2, 32);
      __syncthreads();
      for (int e = tid; e < 1024; e += blockDim.x) sT1[e] = sVV[e] - sT1[e];
      __syncthreads();
      lds_gemm<false, false>(Jt, 32, sT1, 32, sT2, 32, 2, 2, 32);
      __syncthreads();
      for (int e = tid; e < 1024; e += blockDim.x) sT2[e] += sVmU[e];
      __syncthreads();
      lds_gemm<false, true>(sT2, 32, Jp, 32, sVV, 32, 2, 2, 32);
      __syncthreads();
      for (int e = tid; e < 1024; e += blockDim.x)
        out[OUT_VMT1 + e * NOBS + (t - 1)] = sVV[e];
    }

    // ---- advance carries ----
    for (int e = tid; e < 1024; e += blockDim.x) sVnext[e] = sVt[e];
    if (tid < 32) szn[tid] = szt[tid];
    __syncthreads();
    float* tmp = Jt; Jt = Jp; Jp = tmp;
  }
}

// ===========================================================================
extern "C" void kernel_launch(void* const* d_in, const int* in_sizes, int n_in,
                              void* d_out, int out_size, void* d_ws, size_t ws_size,
                              hipStream_t stream)
{
  (void)in_sizes; (void)n_in; (void)out_size; (void)ws_size;
  const float* Y  = (const float*)d_in[0];   // (64, 8192)
  const float* A  = (const float*)d_in[1];   // (32, 32)
  const float* C  = (const float*)d_in[2];   // (64, 32)
  const float* Q  = (const float*)d_in[3];   // (32, 32)
  const float* R  = (const float*)d_in[4];   // (64, 64) diagonal
  const float* Z0 = (const float*)d_in[5];   // (32,)
  const float* V0 = (const float*)d_in[6];   // (32, 32)
  float* out = (float*)d_out;
  float* ws  = (float*)d_ws;                 // needs ~98.1 MB

  kf_preprocess<<<dim3(1), dim3(256), 0, stream>>>(A, Q, R, V0, ws);
  kf_forward  <<<dim3(1), dim3(256), 0, stream>>>(Y, C, Z0, ws, out);
  kf_gain     <<<dim3(NOBS), dim3(128), 0, stream>>>(ws);
  kf_backward <<<dim3(1), dim3(256), 0, stream>>>(C, Z0, ws, out);
}